// Model_57921928954284
// MI455X (gfx1250) — compile-verified
//
#include <hip/hip_runtime.h>

// ---------------------------------------------------------------------------
// GNN: 3 x (gather -> conv1_k3 -> scatter-max -> conv2_k3 -> relu),
// then per-node max over T, then [64,1000] x [1000,3] GEMM via WMMA f32.
// ---------------------------------------------------------------------------

#define N_NODES   64000
#define VERTICES  1000
#define T_DIM     64
#define N_EDGES   1024000
#define N_LAYERS  3
#define BATCH     64

typedef __attribute__((ext_vector_type(2))) float v2f;
typedef __attribute__((ext_vector_type(8))) float v8f;

// Order-preserving IEEE-754 float -> uint key so scatter-max lowers to a
// single native global_atomic_max_u32 (exact & order-independent).
static __device__ __forceinline__ unsigned enc_f32(float f) {
  unsigned u = __float_as_uint(f);
  return (u & 0x80000000u) ? ~u : (u | 0x80000000u);
}
static __device__ __forceinline__ float dec_f32(unsigned k) {
  return __uint_as_float((k & 0x80000000u) ? (k & 0x7fffffffu) : ~k);
}
#define NEG_INF_KEY 0x007FFFFFu   // enc_f32(-inf): empty-segment sentinel

// ---------------------------------------------------------------------------
// Fill aggregation buffer with enc(-inf). 4,096,000 uints as uint4 stores.
// ---------------------------------------------------------------------------
__global__ void init_aggr_kernel(unsigned* __restrict__ aggr) {
  int i = blockIdx.x * blockDim.x + threadIdx.x;
  ((uint4*)aggr)[i] = make_uint4(NEG_INF_KEY, NEG_INF_KEY, NEG_INF_KEY, NEG_INF_KEY);
}

// ---------------------------------------------------------------------------
// One wave per edge: gather x[src] (64 floats = 256B coalesced, float2/lane),
// 3-tap conv with zero padding (halo via wave32 shuffles), atomic-max scatter
// into aggr[dst].  Everything is L2-resident (x = 16.4MB << 192MB L2).
// ---------------------------------------------------------------------------
__global__ void edge_msg_kernel(const float* __restrict__ x,
                                const int*   __restrict__ ei,
                                const float* __restrict__ msg_w,
                                const float* __restrict__ msg_b,
                                unsigned*    __restrict__ aggr,
                                int layer) {
  int gtid = blockIdx.x * blockDim.x + threadIdx.x;
  int edge = gtid >> 5;
  int lane = threadIdx.x & 31;
  if (edge >= N_EDGES) return;

  int src = ei[edge];
  int dst = ei[N_EDGES + edge];

  float w0 = msg_w[layer * 3 + 0];
  float w1 = msg_w[layer * 3 + 1];
  float w2 = msg_w[layer * 3 + 2];
  float b  = msg_b[layer];

  // lane handles t = 2*lane and 2*lane+1
  float2 v = ((const float2*)(x + (size_t)src * T_DIM))[lane];

  float left  = __shfl_up(v.y, 1, 32);   // x[2*lane - 1]
  if (lane == 0) left = 0.0f;
  float right = __shfl_down(v.x, 1, 32); // x[2*lane + 2]
  if (lane == 31) right = 0.0f;

  float m0 = fmaf(w0, left, fmaf(w1, v.x, fmaf(w2, v.y,   b)));
  float m1 = fmaf(w0, v.x,  fmaf(w1, v.y, fmaf(w2, right, b)));

  unsigned* arow = aggr + (size_t)dst * T_DIM + 2 * lane;
  atomicMax(arow,     enc_f32(m0));
  atomicMax(arow + 1, enc_f32(m1));
}

// ---------------------------------------------------------------------------
// One wave per node: out[t] = relu(b + sum_k w0k*x[t+k-1] + w1k*aggr[t+k-1]),
// with empty-segment aggr (-inf sentinel) treated as 0 per the reference.
// ---------------------------------------------------------------------------
__global__ void update_kernel(const float*    __restrict__ xin,
                              const unsigned* __restrict__ aggr,
                              const float*    __restrict__ upd_w,
                              const float*    __restrict__ upd_b,
                              float*          __restrict__ xout,
                              int layer) {
  int gtid = blockIdx.x * blockDim.x + threadIdx.x;
  int node = gtid >> 5;
  int lane = threadIdx.x & 31;
  if (node >= N_NODES) return;

  const float* wl = upd_w + layer * 6;   // [2][3]
  float a0 = wl[0], a1 = wl[1], a2 = wl[2];
  float c0 = wl[3], c1 = wl[4], c2 = wl[5];
  float b  = upd_b[layer];

  float2 xv = ((const float2*)(xin + (size_t)node * T_DIM))[lane];
  uint2  ak = ((const uint2*)(aggr + (size_t)node * T_DIM))[lane];
  float2 av;
  av.x = (ak.x == NEG_INF_KEY) ? 0.0f : dec_f32(ak.x);
  av.y = (ak.y == NEG_INF_KEY) ? 0.0f : dec_f32(ak.y);

  float xl = __shfl_up(xv.y, 1, 32);   if (lane == 0)  xl = 0.0f;
  float xr = __shfl_down(xv.x, 1, 32); if (lane == 31) xr = 0.0f;
  float al = __shfl_up(av.y, 1, 32);   if (lane == 0)  al = 0.0f;
  float ar = __shfl_down(av.x, 1, 32); if (lane == 31) ar = 0.0f;

  float o0 = b;
  o0 = fmaf(a0, xl,   o0); o0 = fmaf(a1, xv.x, o0); o0 = fmaf(a2, xv.y, o0);
  o0 = fmaf(c0, al,   o0); o0 = fmaf(c1, av.x, o0); o0 = fmaf(c2, av.y, o0);
  float o1 = b;
  o1 = fmaf(a0, xv.x, o1); o1 = fmaf(a1, xv.y, o1); o1 = fmaf(a2, xr,   o1);
  o1 = fmaf(c0, av.x, o1); o1 = fmaf(c1, av.y, o1); o1 = fmaf(c2, ar,   o1);

  ((float2*)(xout + (size_t)node * T_DIM))[lane] = make_float2(fmaxf(o0, 0.0f),
                                                               fmaxf(o1, 0.0f));
}

// ---------------------------------------------------------------------------
// One wave per node: max over T=64 via butterfly reduction.
// ---------------------------------------------------------------------------
__global__ void node_max_kernel(const float* __restrict__ x,
                                float* __restrict__ nmax) {
  int gtid = blockIdx.x * blockDim.x + threadIdx.x;
  int node = gtid >> 5;
  int lane = threadIdx.x & 31;
  if (node >= N_NODES) return;
  float2 v = ((const float2*)(x + (size_t)node * T_DIM))[lane];
  float m = fmaxf(v.x, v.y);
  #pragma unroll
  for (int off = 16; off > 0; off >>= 1)
    m = fmaxf(m, __shfl_xor(m, off, 32));
  if (lane == 0) nmax[node] = m;
}

// ---------------------------------------------------------------------------
// Final projection: [64,1000] x [1000,3] + bias via V_WMMA_F32_16X16X4_F32.
// 4 blocks of exactly one wave each (EXEC all-1s, no branches around WMMA).
// Tile = 16 batch rows; N padded 3->16; K=1000 in 250 chunks of 4.
// ISA layouts (7.12.2): A: lane m=lane&15 holds K=2h,2h+1 (h=lane>>4) in v0,v1.
//                       B: lane n=lane&15 holds rows K=2h,2h+1 in v0,v1.
//                       D: vgpr i, lane -> D[i + 8h][n].
// ---------------------------------------------------------------------------
__global__ void out_gemm_kernel(const float* __restrict__ nmax,
                                const float* __restrict__ out_w,
                                const float* __restrict__ out_b,
                                float* __restrict__ out) {
  int lane = threadIdx.x;            // 0..31, one wave per block
  int m    = lane & 15;
  int h    = lane >> 4;              // 0 or 1
  int tile = blockIdx.x;             // 0..3 -> batch rows [16*tile, 16*tile+16)
  int n    = lane & 15;              // output column (valid if n < 3)
  bool nvalid = (n < 3);
  int  nc = nvalid ? n : 0;          // clamped column for safe loads

  const float* arow = nmax + (size_t)(tile * 16 + m) * VERTICES;
  const float* bcol = out_w + (size_t)nc * VERTICES;

  v8f c = {};
  for (int k0 = 0; k0 < VERTICES; k0 += 4) {
    v2f a, b;
    a.x = arow[k0 + 2 * h];
    a.y = arow[k0 + 2 * h + 1];
    float bw0 = bcol[k0 + 2 * h];
    float bw1 = bcol[k0 + 2 * h + 1];
    b.x = nvalid ? bw0 : 0.0f;       // v_cndmask, no divergence
    b.y = nvalid ? bw1 : 0.0f;
    c = __builtin_amdgcn_wmma_f32_16x16x4_f32(
        /*neg_a=*/false, a, /*neg_b=*/false, b,
        /*c_mod=*/(short)0, c, /*reuse_a=*/false, /*reuse_b=*/false);
  }

  float bias = out_b[nc];
  if (nvalid) {
    #pragma unroll
    for (int i = 0; i < 8; ++i) {
      int r = tile * 16 + i + 8 * h;       // batch row
      out[r * 3 + n] = c[i] + bias;
    }
  }
}

// ---------------------------------------------------------------------------
// Host launcher. Inputs (setup_inputs order):
//   0:x [64000,64] f32   1:edge_index [2,1024000] i32
//   2:msg_w [3,3]        3:msg_b [3]
//   4:upd_w [3,2,3]      5:upd_b [3]
//   6:out_w [3,1000]     7:out_b [3]
// Output: [64,3] f32. Workspace: xA, xB (ping-pong), aggr keys, node max.
// ---------------------------------------------------------------------------
extern "C" void kernel_launch(void* const* d_in, const int* in_sizes, int n_in,
                              void* d_out, int out_size, void* d_ws, size_t ws_size,
                              hipStream_t stream) {
  const float* x0    = (const float*)d_in[0];
  const int*   ei    = (const int*)  d_in[1];
  const float* msg_w = (const float*)d_in[2];
  const float* msg_b = (const float*)d_in[3];
  const float* upd_w = (const float*)d_in[4];
  const float* upd_b = (const float*)d_in[5];
  const float* out_w = (const float*)d_in[6];
  const float* out_b = (const float*)d_in[7];
  float* out = (float*)d_out;

  const size_t NT = (size_t)N_NODES * T_DIM;    // 4,096,000 elements
  float*    xA   = (float*)d_ws;
  float*    xB   = xA + NT;
  unsigned* aggr = (unsigned*)(xB + NT);
  float*    nmax = (float*)(aggr + NT);

  const int BLK = 256;
  const int init_blocks = (int)(NT / 4 / BLK);            // 4000
  const int edge_blocks = (N_EDGES * 32) / BLK;           // 128000
  const int node_blocks = (N_NODES * 32) / BLK;           // 8000

  const float* xin = x0;
  float* bufs[N_LAYERS] = { xA, xB, xA };
  for (int l = 0; l < N_LAYERS; ++l) {
    init_aggr_kernel<<<init_blocks, BLK, 0, stream>>>(aggr);
    edge_msg_kernel <<<edge_blocks, BLK, 0, stream>>>(xin, ei, msg_w, msg_b, aggr, l);
    update_kernel   <<<node_blocks, BLK, 0, stream>>>(xin, aggr, upd_w, upd_b, bufs[l], l);
    xin = bufs[l];
  }
  node_max_kernel<<<node_blocks, BLK, 0, stream>>>(xin, nmax);
  out_gemm_kernel<<<BATCH / 16, 32, 0, stream>>>(nmax, out_w, out_b, out);
}